// SimpleGNN_38603166056972
// MI455X (gfx1250) — compile-verified
//
#include <hip/hip_runtime.h>
#include <hip/hip_bf16.h>

typedef __attribute__((ext_vector_type(2))) float v2f;
typedef __attribute__((ext_vector_type(8))) float v8f;

static __device__ __forceinline__ int imin(int a, int b) { return a < b ? a : b; }

// ---------------- elementwise / setup kernels ----------------

__global__ void fill_f32(float* __restrict__ p, float v, int n) {
  int i = blockIdx.x * blockDim.x + threadIdx.x;
  if (i < n) p[i] = v;
}

// deg[col[e]] += ew[e]   (deg pre-filled with 1.0 == self-loop weight)
__global__ void deg_accum(float* __restrict__ deg, const int* __restrict__ ei,
                          const float* __restrict__ ew, int E) {
  int e = blockIdx.x * blockDim.x + threadIdx.x;
  if (e < E) atomicAdd(&deg[ei[E + e]], ew[e]);
}

// in place: d = deg > 0 ? rsqrt(deg) : 0
__global__ void rsqrt_k(float* __restrict__ d, int n) {
  int i = blockIdx.x * blockDim.x + threadIdx.x;
  if (i < n) {
    float v = d[i];
    d[i] = (v > 0.0f) ? rsqrtf(v) : 0.0f;
  }
}

// norm[e] = dis[row]*dis[col]*ew ; self-loops (e>=E): dis[n]^2 * 1
__global__ void norm_k(float* __restrict__ nrm, const float* __restrict__ dis,
                       const int* __restrict__ ei, const float* __restrict__ ew,
                       int E, int N) {
  int e = blockIdx.x * blockDim.x + threadIdx.x;
  if (e >= E + N) return;
  float v;
  if (e < E) {
    v = dis[ei[e]] * dis[ei[E + e]] * ew[e];
  } else {
    float d = dis[e - E];
    v = d * d;
  }
  nrm[e] = v;
}

// buf[i] = act(buf[i] + bias[i & dmask])
__global__ void bias_act(float* __restrict__ buf, const float* __restrict__ bias,
                         int total, int dmask, int relu) {
  int i = blockIdx.x * blockDim.x + threadIdx.x;
  if (i >= total) return;
  float v = buf[i] + bias[i & dmask];
  if (relu) v = fmaxf(v, 0.0f);
  buf[i] = v;
}

// ---------------- WMMA fp32 GEMM:  H[N x Dout] = X[N x K] @ W[K x Dout] ----------------
// blockIdx.x = 16-row tile; blockDim.x = 32*(Dout/16); wave w owns column tile w.
// All waves in a block share the same 16xK A tile, so it is staged ONCE into LDS
// (row stride 132 floats: keeps 16 A-lanes on distinct banks (4*lm mod 64),
// float4 stores 16B-aligned, ds_load_b64 A-frag reads 8B-aligned).
// A frag (16x4 f32, ISA 7.12.2): lanes 0-15 hold {K=k,k+1}, lanes 16-31 {K=k+2,k+3}.
// C/D (16x16): VGPR v: lanes 0-15 -> M=v, lanes 16-31 -> M=v+8; N = lane&15.
#define TILE_LDS_STRIDE 132
#define KMAX 128

__global__ void gemm_wmma_f32(float* __restrict__ H, const float* __restrict__ X,
                              const float* __restrict__ W, int n_rows, int K, int Dout) {
  __shared__ float tile[16 * TILE_LDS_STRIDE];

  const int tid   = threadIdx.x;
  const int lane  = tid & 31;
  const int wave  = tid >> 5;
  const int m0    = blockIdx.x * 16;
  const int n0    = wave * 16;
  const int lm    = lane & 15;          // M (for A) / N (for B, C)
  const int khalf = (lane >> 4) * 2;    // 0 or 2

  // --- cooperative stage of the 16 x K A tile into LDS (float4 granularity) ---
  const int nvec = 16 * (K >> 2);       // float4 slots
  const int vpr  = K >> 2;              // float4 per row
  for (int i = tid; i < nvec; i += blockDim.x) {
    const int r  = i / vpr;
    const int c4 = (i - r * vpr) << 2;
    const int gr = imin(m0 + r, n_rows - 1);   // clamp: no divergence later
    const float4 v = *reinterpret_cast<const float4*>(X + (size_t)gr * K + c4);
    *reinterpret_cast<float4*>(&tile[r * TILE_LDS_STRIDE + c4]) = v;
  }
  __syncthreads();

  const float* __restrict__ tA   = &tile[lm * TILE_LDS_STRIDE];
  const float* __restrict__ wcol = W + n0 + lm;

  v8f c = {};
  for (int k = 0; k < K; k += 4) {
    v2f a, b;
    a.x = tA[k + khalf];
    a.y = tA[k + khalf + 1];
    b.x = wcol[(size_t)(k + khalf) * Dout];
    b.y = wcol[(size_t)(k + khalf + 1) * Dout];
    c = __builtin_amdgcn_wmma_f32_16x16x4_f32(false, a, false, b, (short)0, c,
                                              false, false);
  }

  const int mbase = m0 + ((lane >> 4) << 3);
#pragma unroll
  for (int v = 0; v < 8; ++v) {
    int m = mbase + v;
    if (m < n_rows) H[(size_t)m * Dout + n0 + lm] = c[v];
  }
}

// ---------------- edge aggregation: out[col] += h[row] * norm  ----------------
// One wave per edge; each lane handles 4 contiguous features (float4 load + 4 f32 atomics).
__global__ void aggregate_k(float* __restrict__ out, const float* __restrict__ h,
                            const float* __restrict__ nrm, const int* __restrict__ ei,
                            int E, int N, int D) {
  const int wgl  = blockIdx.x * (blockDim.x >> 5) + (threadIdx.x >> 5);
  const int lane = threadIdx.x & 31;
  if (wgl >= E + N) return;

  int row, col;
  if (wgl < E) {
    row = ei[wgl];
    col = ei[E + wgl];
  } else {
    row = wgl - E;
    col = row;
  }
  const float s = nrm[wgl];

  const int f = lane * 4;
  if (f < D) {
    const float4 v = *reinterpret_cast<const float4*>(h + (size_t)row * D + f);
    float* dst = out + (size_t)col * D + f;
    atomicAdd(dst + 0, v.x * s);
    atomicAdd(dst + 1, v.y * s);
    atomicAdd(dst + 2, v.z * s);
    atomicAdd(dst + 3, v.w * s);
  }
}

// ---------------- driver ----------------

extern "C" void kernel_launch(void* const* d_in, const int* in_sizes, int n_in,
                              void* d_out, int out_size, void* d_ws, size_t ws_size,
                              hipStream_t stream) {
  const float* x  = (const float*)d_in[0];
  const int*   ei = (const int*)  d_in[1];
  const float* ew = (const float*)d_in[2];
  const float* w0 = (const float*)d_in[3];
  const float* b0 = (const float*)d_in[4];
  const float* w1 = (const float*)d_in[5];
  const float* b1 = (const float*)d_in[6];
  const float* w2 = (const float*)d_in[7];
  const float* b2 = (const float*)d_in[8];

  const int DIN  = 128;
  const int DH   = 128;
  const int DOUT = in_sizes[8];            // 64
  const int N    = in_sizes[0] / DIN;      // 100000
  const int E    = in_sizes[2];            // 1600000
  const int T    = E + N;                  // edges incl. self-loops

  float* dis  = (float*)d_ws;              // N
  float* nrm  = dis + N;                   // T
  float* bufA = nrm + T;                   // N*DH
  float* bufB = bufA + (size_t)N * DH;     // N*DH
  float* out  = (float*)d_out;             // N*DOUT

  auto cdiv = [](int a, int b) { return (a + b - 1) / b; };

  // --- degree / norm precompute (shared by all 3 layers) ---
  fill_f32 <<<cdiv(N, 256), 256, 0, stream>>>(dis, 1.0f, N);
  deg_accum<<<cdiv(E, 256), 256, 0, stream>>>(dis, ei, ew, E);
  rsqrt_k  <<<cdiv(N, 256), 256, 0, stream>>>(dis, N);
  norm_k   <<<cdiv(T, 256), 256, 0, stream>>>(nrm, dis, ei, ew, E, N);

  const int rowTiles = cdiv(N, 16);

  // --- layer 0: h = relu(scatter(gather(x@w0)) + b0) ---
  gemm_wmma_f32<<<rowTiles, 32 * (DH / 16), 0, stream>>>(bufA, x, w0, N, DIN, DH);
  fill_f32     <<<cdiv(N * DH, 256), 256, 0, stream>>>(bufB, 0.0f, N * DH);
  aggregate_k  <<<cdiv(T, 8), 256, 0, stream>>>(bufB, bufA, nrm, ei, E, N, DH);
  bias_act     <<<cdiv(N * DH, 256), 256, 0, stream>>>(bufB, b0, N * DH, DH - 1, 1);

  // --- layer 1 ---
  gemm_wmma_f32<<<rowTiles, 32 * (DH / 16), 0, stream>>>(bufA, bufB, w1, N, DH, DH);
  fill_f32     <<<cdiv(N * DH, 256), 256, 0, stream>>>(bufB, 0.0f, N * DH);
  aggregate_k  <<<cdiv(T, 8), 256, 0, stream>>>(bufB, bufA, nrm, ei, E, N, DH);
  bias_act     <<<cdiv(N * DH, 256), 256, 0, stream>>>(bufB, b1, N * DH, DH - 1, 1);

  // --- layer 2 (no relu), aggregate directly into d_out ---
  gemm_wmma_f32<<<rowTiles, 32 * (DOUT / 16), 0, stream>>>(bufA, bufB, w2, N, DH, DOUT);
  fill_f32     <<<cdiv(N * DOUT, 256), 256, 0, stream>>>(out, 0.0f, N * DOUT);
  aggregate_k  <<<cdiv(T, 8), 256, 0, stream>>>(out, bufA, nrm, ei, E, N, DOUT);
  bias_act     <<<cdiv(N * DOUT, 256), 256, 0, stream>>>(out, b2, N * DOUT, DOUT - 1, 0);
}